// HairBundleSDE_83597243449822
// MI455X (gfx1250) — compile-verified
//
#include <hip/hip_runtime.h>
#include <hip/hip_bf16.h>

typedef __attribute__((ext_vector_type(4))) float v4f;

namespace hb {
// HairBundle model constants
constexpr float K_GS      = 0.75f;
constexpr float K_SP      = 0.6f;
constexpr float K_ES      = 0.45f;
constexpr float D_GATE    = 0.5f;
constexpr float INV_DELTA = 4.0f;    // 1 / 0.25
constexpr float F_MAX     = 0.35f;
constexpr float S_CA      = 0.9f;
constexpr float C_M       = 1.2f;
constexpr float K_M       = 0.8f;
constexpr float C_GS      = 0.7f;
constexpr float K_GSR     = 0.5f;
constexpr float C_T       = 0.3f;
constexpr float K_T       = 0.4f;
constexpr float INV_LAM_A = 0.1f;    // 1 / 10
constexpr float HB_NOISE  = 0.05f;
constexpr float A_NOISE   = 0.02f;
constexpr float F_AMP     = 0.5f;
constexpr float F_OMEGA   = 6.283185307179586f;
} // namespace hb

// drift of one row; p_o via one v_exp_f32 (TRANS, co-executes with VALU)
__device__ __forceinline__ void hb_row(float x_hb, float x_a, float p_m,
                                       float p_gs, float p_t, float force,
                                       float& d0, float& d1, float& d2,
                                       float& d3, float& d4) {
  using namespace hb;
  const float z    = INV_DELTA * (x_hb - x_a);
  const float p_o  = 1.0f / (1.0f + __expf(-z));
  const float f_gs = K_GS * (x_hb - x_a - D_GATE * p_o);
  d0 = force - f_gs - K_SP * x_hb;
  d1 = INV_LAM_A * (f_gs - K_ES * x_a - F_MAX * (1.0f - S_CA * p_m));
  d2 = C_M  * p_o * (1.0f - p_m ) - K_M   * p_m;
  d3 = C_GS * p_o * (1.0f - p_gs) - K_GSR * p_gs;
  d4 = C_T  * p_o * (1.0f - p_t ) - K_T   * p_t;
}

// s_wait_asynccnt <= N (ASYNCcnt, gfx1250 split dependency counters)
#define HB_ASYNC_WAIT(N) asm volatile("s_wait_asynccnt " N ::: "memory")

// Async-DMA one 80B per-thread chunk (5 x B128) from global to LDS.
// ISA: IOFFSET is added to BOTH the LDS (VDST) and global (VADDR) address,
// so one address pair covers all five transfers.
__device__ __forceinline__ void hb_async_load_80B(unsigned lds_addr,
                                                  unsigned long long gaddr) {
  asm volatile(
      "global_load_async_to_lds_b128 %0, %1, off\n\t"
      "global_load_async_to_lds_b128 %0, %1, off offset:16\n\t"
      "global_load_async_to_lds_b128 %0, %1, off offset:32\n\t"
      "global_load_async_to_lds_b128 %0, %1, off offset:48\n\t"
      "global_load_async_to_lds_b128 %0, %1, off offset:64"
      :
      : "v"(lds_addr), "v"(gaddr)
      : "memory");
}

constexpr int kThreads = 256;  // 8 wave32 waves per block

// Main kernel: per block, double-buffered async staging of 20KB tiles
// (256 threads x 80B = 256 groups x 4 rows). Thread tid owns LDS slice
// smem[buf][tid*5 .. tid*5+4]; no cross-thread sharing => no barriers,
// only s_wait_asynccnt ordering. Persistent grid-stride over tiles.
__global__ __launch_bounds__(kThreads) void hb_sde_async(
    const float* __restrict__ tptr, const float* __restrict__ x,
    float* __restrict__ out, const long long ntiles, const long long batch) {
  using namespace hb;
  __shared__ v4f smem[2][kThreads * 5];  // 2 x 20KB = 40KB per block

  const float force = F_AMP * __sinf(F_OMEGA * tptr[0]);  // phase = 0
  const unsigned tid = threadIdx.x;
  const long long tileStride = gridDim.x;
  const unsigned long long xbase = (unsigned long long)(uintptr_t)x;

  const v4f* curbuf = &smem[0][tid * 5];
  const v4f* nxtbuf = &smem[1][tid * 5];
  unsigned curlds = (unsigned)(uintptr_t)curbuf;  // low 32 bits = LDS offset
  unsigned nxtlds = (unsigned)(uintptr_t)nxtbuf;

  v4f* __restrict__ dr = (v4f*)out;                  // drift     [batch,5]
  v4f* __restrict__ df = (v4f*)(out + 5ll * batch);  // diffusion [batch,5]

  const v4f D0 = {HB_NOISE, A_NOISE, 0.0f, 0.0f};
  const v4f D1 = {0.0f, HB_NOISE, A_NOISE, 0.0f};
  const v4f D2 = {0.0f, 0.0f, HB_NOISE, A_NOISE};
  const v4f D3 = {0.0f, 0.0f, 0.0f, HB_NOISE};
  const v4f D4 = {A_NOISE, 0.0f, 0.0f, 0.0f};

  long long T = blockIdx.x;
  if (T < ntiles) {  // prologue: stage first tile
    hb_async_load_80B(
        curlds, xbase + 80ull * (unsigned long long)(T * kThreads + tid));
  }

  for (; T < ntiles; T += tileStride) {
    const long long nextT = T + tileStride;
    if (nextT < ntiles) {
      // stage next tile into the other buffer, then wait for current tile:
      // async ops complete in order, so ASYNCcnt<=5 => first 5 are done.
      hb_async_load_80B(
          nxtlds, xbase + 80ull * (unsigned long long)(nextT * kThreads + tid));
      HB_ASYNC_WAIT("5");
    } else {
      HB_ASYNC_WAIT("0");
    }

    // read back own 80B slice from LDS (ds_load_b128 x5)
    const v4f v0 = curbuf[0];
    const v4f v1 = curbuf[1];
    const v4f v2 = curbuf[2];
    const v4f v3 = curbuf[3];
    const v4f v4 = curbuf[4];

    // rows: r0=(v0.x..v1.x) r1=(v1.y..v2.y) r2=(v2.z..v3.z) r3=(v3.w..v4.w)
    float o0, o1, o2, o3, o4, o5, o6, o7, o8, o9;
    float o10, o11, o12, o13, o14, o15, o16, o17, o18, o19;

    hb_row(v0.x, v0.y, v0.z, v0.w, v1.x, force, o0,  o1,  o2,  o3,  o4);
    hb_row(v1.y, v1.z, v1.w, v2.x, v2.y, force, o5,  o6,  o7,  o8,  o9);
    hb_row(v2.z, v2.w, v3.x, v3.y, v3.z, force, o10, o11, o12, o13, o14);
    hb_row(v3.w, v4.x, v4.y, v4.z, v4.w, force, o15, o16, o17, o18, o19);

    const v4f w0 = {o0,  o1,  o2,  o3};
    const v4f w1 = {o4,  o5,  o6,  o7};
    const v4f w2 = {o8,  o9,  o10, o11};
    const v4f w3 = {o12, o13, o14, o15};
    const v4f w4 = {o16, o17, o18, o19};

    const long long b = 5ll * (T * kThreads + tid);  // float4 index

    __builtin_nontemporal_store(w0, dr + b + 0);
    __builtin_nontemporal_store(w1, dr + b + 1);
    __builtin_nontemporal_store(w2, dr + b + 2);
    __builtin_nontemporal_store(w3, dr + b + 3);
    __builtin_nontemporal_store(w4, dr + b + 4);

    __builtin_nontemporal_store(D0, df + b + 0);
    __builtin_nontemporal_store(D1, df + b + 1);
    __builtin_nontemporal_store(D2, df + b + 2);
    __builtin_nontemporal_store(D3, df + b + 3);
    __builtin_nontemporal_store(D4, df + b + 4);

    // swap double buffers
    const v4f* tb = curbuf; curbuf = nxtbuf; nxtbuf = tb;
    const unsigned tl = curlds; curlds = nxtlds; nxtlds = tl;
  }
}

// Cleanup: scalar per-row for the <=1023 rows not covered by full tiles
// plus any batch%4 remainder (512 rows for batch = 8M).
__global__ __launch_bounds__(256) void hb_sde_cleanup(
    const float* __restrict__ tptr, const float* __restrict__ x,
    float* __restrict__ out, const long long start_row, const long long nrows,
    const long long batch) {
  using namespace hb;
  const long long i =
      start_row + (long long)blockIdx.x * blockDim.x + threadIdx.x;
  if (i >= start_row + nrows) return;
  const float force = F_AMP * __sinf(F_OMEGA * tptr[0]);
  const float* r = x + 5ll * i;
  float d0, d1, d2, d3, d4;
  hb_row(r[0], r[1], r[2], r[3], r[4], force, d0, d1, d2, d3, d4);
  float* dro = out + 5ll * i;
  float* dfo = out + 5ll * batch + 5ll * i;
  dro[0] = d0; dro[1] = d1; dro[2] = d2; dro[3] = d3; dro[4] = d4;
  dfo[0] = HB_NOISE; dfo[1] = A_NOISE; dfo[2] = 0.0f; dfo[3] = 0.0f;
  dfo[4] = 0.0f;
}

extern "C" void kernel_launch(void* const* d_in, const int* in_sizes, int n_in,
                              void* d_out, int out_size, void* d_ws,
                              size_t ws_size, hipStream_t stream) {
  const float* t = (const float*)d_in[0];  // [1]
  const float* x = (const float*)d_in[1];  // [batch, 5] row-major f32
  float* out = (float*)d_out;  // drift [batch,5] ++ diffusion [batch,5]

  const long long total   = (long long)in_sizes[1];  // batch * 5
  const long long batch   = total / 5;
  const long long ngroups = batch / 4;               // 4 rows per thread
  const long long ntiles  = ngroups / kThreads;      // 1024 rows per tile

  if (ntiles > 0) {
    const int grid = (int)(ntiles < 4096 ? ntiles : 4096);
    hipLaunchKernelGGL(hb_sde_async, dim3(grid), dim3(kThreads), 0, stream,
                       t, x, out, ntiles, batch);
  }

  const long long rows_main = ntiles * (4ll * kThreads);
  const long long rem_rows  = batch - rows_main;  // < 1024 + 3
  if (rem_rows > 0) {
    const int cgrid = (int)((rem_rows + 255) / 256);
    hipLaunchKernelGGL(hb_sde_cleanup, dim3(cgrid), dim3(256), 0, stream,
                       t, x, out, rows_main, rem_rows, batch);
  }
}